// GraphSAGEBackbone_44805098832144
// MI455X (gfx1250) — compile-verified
//
#include <hip/hip_runtime.h>

typedef __attribute__((ext_vector_type(2))) float v2f;
typedef __attribute__((ext_vector_type(8))) float v8f;

#define FEAT 128          // D
#define TPAD 132          // LDS pitch (floats): breaks bank conflicts, keeps 8B align

// ---------------------------------------------------------------- zero fill
__global__ __launch_bounds__(256) void zero_f32(float* __restrict__ p, long long n) {
    long long i = (long long)blockIdx.x * blockDim.x + threadIdx.x;
    if (i < n) p[i] = 0.0f;
}

// ---------------------------------------------------------------- in-degree
__global__ __launch_bounds__(256) void count_deg(const long long* __restrict__ dst,
                                                 float* __restrict__ cnt, int E) {
    int e = blockIdx.x * blockDim.x + threadIdx.x;
    if (e < E) atomicAdd(&cnt[dst[e]], 1.0f);
}

// -------------------------------------------------- edge scatter: one wave/edge
// lane l handles features [4l .. 4l+3]: float4 gather + 4 L2-resident atomics
__global__ __launch_bounds__(256) void scatter_add(const float* __restrict__ h,
                                                   const long long* __restrict__ src,
                                                   const long long* __restrict__ dst,
                                                   float* __restrict__ agg, int E) {
    long long g = (long long)blockIdx.x * blockDim.x + threadIdx.x;
    int e = (int)(g >> 5);
    if (e >= E) return;
    int lane = (int)(g & 31);
    long long s = src[e];
    long long d = dst[e];
    const float4 v = *(const float4*)(h + s * FEAT + lane * 4);
    float* a = agg + d * FEAT + lane * 4;
    atomicAdd(a + 0, v.x);
    atomicAdd(a + 1, v.y);
    atomicAdd(a + 2, v.z);
    atomicAdd(a + 3, v.w);
}

// -------------------------- fused SAGE layer GEMM: relu(agg/deg @ Wl + h @ Wr + b)
// Block = 8 waves; block owns rows [row0, row0+16); wave w owns cols [16w, 16w+16).
// fp32 WMMA 16x16x4: A frag = v2f (lane: m = lane&15, k = 2*(lane>>4) + {0,1})
//                    B frag = v2f (lane: n = lane&15, k = 2*(lane>>4) + {0,1})
//                    C tile = v8f (lane,v: m = v + 8*(lane>>4), n = lane&15)
__global__ __launch_bounds__(256) void sage_gemm(const float* __restrict__ h,
                                                 const float* __restrict__ agg,
                                                 const float* __restrict__ cnt,
                                                 const float* __restrict__ Wl,
                                                 const float* __restrict__ Wr,
                                                 const float* __restrict__ bias,
                                                 float* __restrict__ out, int n_rows) {
    __shared__ float sAgg[16 * TPAD];
    __shared__ float sH[16 * TPAD];

    const int row0 = blockIdx.x * 16;
    if (row0 >= n_rows) return;
    const int tid = threadIdx.x;

    // Stage 16x128 panels of h and normalized agg into LDS (512 float4 loads).
    for (int i = tid; i < 16 * 32; i += 256) {
        int r = i >> 5;            // row within panel
        int c = (i & 31) << 2;     // feature column (float4 granularity)
        int gr = row0 + r;
        if (gr >= n_rows) gr = n_rows - 1;   // clamp (unused rows never stored)
        float4 hv = *(const float4*)(h + (size_t)gr * FEAT + c);
        float4 av = *(const float4*)(agg + (size_t)gr * FEAT + c);
        float rd = 1.0f / fmaxf(cnt[gr], 1.0f);
        av.x *= rd; av.y *= rd; av.z *= rd; av.w *= rd;
        *(float4*)(sH + r * TPAD + c) = hv;
        *(float4*)(sAgg + r * TPAD + c) = av;
    }
    __syncthreads();

    const int wave = tid >> 5;
    const int lane = tid & 31;
    const int n0 = wave * 16;
    const int m = lane & 15;
    const int kb = (lane >> 4) << 1;   // 0 or 2
    const int n = n0 + (lane & 15);

    v8f acc = {};
#pragma unroll 8
    for (int k0 = 0; k0 < FEAT; k0 += 4) {
        int k = k0 + kb;
        // agg @ Wl
        v2f a0 = *(const v2f*)(sAgg + m * TPAD + k);
        v2f b0; b0.x = Wl[(size_t)k * FEAT + n]; b0.y = Wl[(size_t)(k + 1) * FEAT + n];
        acc = __builtin_amdgcn_wmma_f32_16x16x4_f32(false, a0, false, b0,
                                                    (short)0, acc, false, false);
        // h @ Wr
        v2f a1 = *(const v2f*)(sH + m * TPAD + k);
        v2f b1; b1.x = Wr[(size_t)k * FEAT + n]; b1.y = Wr[(size_t)(k + 1) * FEAT + n];
        acc = __builtin_amdgcn_wmma_f32_16x16x4_f32(false, a1, false, b1,
                                                    (short)0, acc, false, false);
    }

    const float bv = bias[n];
    const int mbase = (lane >> 4) * 8;
    float* orow = out + (size_t)(row0 + mbase) * FEAT + n;

    if (row0 + 16 <= n_rows) {
        // Fast path (always taken when n_rows % 16 == 0): 8 unguarded coalesced stores.
#pragma unroll
        for (int v = 0; v < 8; ++v) {
            orow[(size_t)v * FEAT] = fmaxf(acc[v] + bv, 0.0f);
        }
    } else {
        // Ragged tail tile.
#pragma unroll
        for (int v = 0; v < 8; ++v) {
            if (row0 + mbase + v < n_rows) {
                orow[(size_t)v * FEAT] = fmaxf(acc[v] + bv, 0.0f);
            }
        }
    }
}

// ---------------------------------------------------------------- launcher
extern "C" void kernel_launch(void* const* d_in, const int* in_sizes, int n_in,
                              void* d_out, int out_size, void* d_ws, size_t ws_size,
                              hipStream_t stream) {
    const float*     x   = (const float*)d_in[0];
    const long long* ei  = (const long long*)d_in[1];
    const float*     Wl0 = (const float*)d_in[2];
    const float*     Wr0 = (const float*)d_in[3];
    const float*     b0  = (const float*)d_in[4];
    const float*     Wl1 = (const float*)d_in[5];
    const float*     Wr1 = (const float*)d_in[6];
    const float*     b1  = (const float*)d_in[7];
    float* out = (float*)d_out;

    const int N = in_sizes[0] / FEAT;
    const int E = in_sizes[1] / 2;
    const long long* src = ei;
    const long long* dst = ei + E;

    // workspace: [cnt: N floats (padded)] [agg: N*FEAT floats]
    float* cnt = (float*)d_ws;
    float* aggbuf = cnt + (((size_t)N + 255) & ~(size_t)255);
    const long long nd = (long long)N * FEAT;

    dim3 blk(256);
    const int gN   = (N + 255) / 256;
    const int gE   = (E + 255) / 256;
    const int gND  = (int)((nd + 255) / 256);
    const int gSct = (int)(((long long)E * 32 + 255) / 256);
    const int gGemm = (N + 15) / 16;

    // degree counts (shared by both layers)
    zero_f32<<<gN, blk, 0, stream>>>(cnt, N);
    count_deg<<<gE, blk, 0, stream>>>(dst, cnt, E);

    // ---- layer 1: x -> out
    zero_f32<<<gND, blk, 0, stream>>>(aggbuf, nd);
    scatter_add<<<gSct, blk, 0, stream>>>(x, src, dst, aggbuf, E);
    sage_gemm<<<gGemm, blk, 0, stream>>>(x, aggbuf, cnt, Wl0, Wr0, b0, out, N);

    // ---- layer 2: out -> out (in-place safe: agg materialized first,
    //      each gemm block stages its own rows to LDS before storing)
    zero_f32<<<gND, blk, 0, stream>>>(aggbuf, nd);
    scatter_add<<<gSct, blk, 0, stream>>>(out, src, dst, aggbuf, E);
    sage_gemm<<<gGemm, blk, 0, stream>>>(out, aggbuf, cnt, Wl1, Wr1, b1, out, N);
}